// Isolator_86406152061201
// MI455X (gfx1250) — compile-verified
//
#include <hip/hip_runtime.h>
#include <math.h>

#define PI_F 3.14159265358979323846f
#define BN_EPS 1e-5f
#define NF 255

typedef __attribute__((ext_vector_type(2))) float v2f;
typedef __attribute__((ext_vector_type(8))) float v8f;

// Branchless clamped row fetch: row[] is a valid channel row of length Tin,
// so clamping the index keeps the load safe; out-of-range taps contribute 0.
__device__ __forceinline__ float fetch_clamped(const float* __restrict__ row,
                                               int pos, int Tin)
{
  int posc = pos < 0 ? 0 : pos;
  posc = posc >= Tin ? Tin - 1 : posc;
  float v = row[posc];
  return (pos == posc) ? v : 0.f;
}

// ---------------------------------------------------------------------------
// Fused conv1d + bias + BN + (optional residual) + ReLU  (scalar fallback,
// used for the skinny early layers where a 16x16 WMMA tile cannot be filled).
// res_mode: 0 = none, 1 = identity residual, 2 = downsample conv + BN residual
// ---------------------------------------------------------------------------
__global__ void conv_layer_scalar(
    const float* __restrict__ X, int cin, int Tin,
    const float* __restrict__ W, const float* __restrict__ Bb,
    int cout, int Tout, int stride, int dil, int pad,
    const float* __restrict__ g, const float* __restrict__ be,
    const float* __restrict__ mn, const float* __restrict__ vr,
    int res_mode, const float* __restrict__ Xres, int cinX, int TinX, int sres,
    const float* __restrict__ dsw, const float* __restrict__ dsg,
    const float* __restrict__ dsb, const float* __restrict__ dsm,
    const float* __restrict__ dsv,
    float* __restrict__ out, int B)
{
  int idx = blockIdx.x * blockDim.x + threadIdx.x;
  int total = B * cout * Tout;
  if (idx >= total) return;
  int t  = idx % Tout;
  int co = (idx / Tout) % cout;
  int b  = idx / (Tout * cout);

  const float* Xb = X + (size_t)b * cin * Tin;
  float acc = Bb[co];
  int base = stride * t - pad;
  for (int ci = 0; ci < cin; ci++) {
    const float* xr = Xb + (size_t)ci * Tin;
    const float* wr = W + ((size_t)co * cin + ci) * 5;
#pragma unroll
    for (int k = 0; k < 5; k++) {
      int pos = base + dil * k;
      acc += wr[k] * fetch_clamped(xr, pos, Tin);
    }
  }
  float sc  = g[co] * rsqrtf(vr[co] + BN_EPS);
  float val = (acc - mn[co]) * sc + be[co];

  if (res_mode == 1) {
    val += Xres[((size_t)b * cout + co) * TinX + t];
  } else if (res_mode == 2) {
    float r = 0.f;
    const float* Xr = Xres + (size_t)b * cinX * TinX + (size_t)sres * t;
    for (int ci = 0; ci < cinX; ci++) r += dsw[co * cinX + ci] * Xr[(size_t)ci * TinX];
    float sd = dsg[co] * rsqrtf(dsv[co] + BN_EPS);
    val += (r - dsm[co]) * sd + dsb[co];
  }
  out[((size_t)b * cout + co) * Tout + t] = fmaxf(val, 0.f);
}

// ---------------------------------------------------------------------------
// Implicit-GEMM conv1d via V_WMMA_F32_16X16X4_F32. One wave computes a
// 16(cout) x 32(t) tile: the A (weight) fragment is loaded once per K-step
// and fed to two WMMAs (two adjacent 16-wide t tiles). K = cin*5, stepped 4
// at a time. The (ci,k) decomposition of the per-lane K index is tracked
// incrementally (no div/mod), and B-element loads are branchless clamp-loads.
//   A (16x4): lane m=lane&15 row of W; K = kb + 2*(lane>>4) + {0,1}
//   B (4x16): lane n=lane&15 column (time); same K-pair striping
//   C (16x16,f32): VGPR i -> row i + 8*(lane>>4), col lane&15
// Epilogue fuses bias + BN + residual + ReLU for both t tiles.
// ---------------------------------------------------------------------------
__global__ void conv_layer_wmma(
    const float* __restrict__ X, int cin, int Tin,
    const float* __restrict__ W, const float* __restrict__ Bb,
    int cout, int Tout, int stride, int dil, int pad,
    const float* __restrict__ g, const float* __restrict__ be,
    const float* __restrict__ mn, const float* __restrict__ vr,
    int res_mode, const float* __restrict__ Xres, int cinX, int TinX, int sres,
    const float* __restrict__ dsw, const float* __restrict__ dsg,
    const float* __restrict__ dsb, const float* __restrict__ dsm,
    const float* __restrict__ dsv,
    float* __restrict__ out, int B)
{
  int lane   = threadIdx.x & 31;
  int tilesT = Tout >> 5;            // 32 t per wave
  int tilesC = cout >> 4;
  int tile   = blockIdx.x;
  int tt = tile % tilesT;
  int cc = (tile / tilesT) % tilesC;
  int b  = tile / (tilesT * tilesC);
  if (b >= B) return;
  int co0 = cc << 4;
  int t0  = tt << 5;

  int ln15 = lane & 15;   // A row (M) / B column (N)
  int pair = lane >> 4;   // selects K pair within 4-wide step
  int Ktot = cin * 5;

  const float* Wp = W + ((size_t)(co0 + ln15)) * Ktot + 2 * pair;
  const float* Xb = X + (size_t)b * cin * Tin;
  int tbase0 = stride * (t0 + ln15) - pad;
  int tbase1 = tbase0 + 16 * stride;

  // incremental (ci, dil*k) tracking for this lane's two K indices
  int q0 = 2 * pair;                 // < 4, so ci = 0, k = q
  int dk0 = dil * q0;
  int dk1 = dil * (q0 + 1);
  const float* row0 = Xb;
  const float* row1 = Xb;
  const int wrap = 5 * dil;

  v8f acc0 = {0.f, 0.f, 0.f, 0.f, 0.f, 0.f, 0.f, 0.f};
  v8f acc1 = {0.f, 0.f, 0.f, 0.f, 0.f, 0.f, 0.f, 0.f};

  for (int kb = 0; kb < Ktot; kb += 4) {
    v2f a;
    a.x = Wp[0];
    a.y = Wp[1];
    Wp += 4;

    v2f bf0, bf1;
    bf0.x = fetch_clamped(row0, tbase0 + dk0, Tin);
    bf0.y = fetch_clamped(row1, tbase0 + dk1, Tin);
    bf1.x = fetch_clamped(row0, tbase1 + dk0, Tin);
    bf1.y = fetch_clamped(row1, tbase1 + dk1, Tin);

    acc0 = __builtin_amdgcn_wmma_f32_16x16x4_f32(
        false, a, false, bf0, (short)0, acc0, false, false);
    acc1 = __builtin_amdgcn_wmma_f32_16x16x4_f32(
        false, a, false, bf1, (short)0, acc1, false, false);

    // advance per-lane K indices by 4 (wraps past one 5-tap group at most once)
    int ndk0 = dk0 + 4 * dil;
    int w0 = (ndk0 >= wrap);
    dk0  = w0 ? ndk0 - wrap : ndk0;
    row0 = w0 ? row0 + Tin : row0;

    int ndk1 = dk1 + 4 * dil;
    int w1 = (ndk1 >= wrap);
    dk1  = w1 ? ndk1 - wrap : ndk1;
    row1 = w1 ? row1 + Tin : row1;
  }

  int N = ln15;
#pragma unroll
  for (int i = 0; i < 8; i++) {
    int M  = i + 8 * pair;
    int co = co0 + M;
    float bb = Bb[co];
    float sc = g[co] * rsqrtf(vr[co] + BN_EPS);
    float mm = mn[co];
    float bt = be[co];
    float sd = 0.f, dm = 0.f, db = 0.f;
    if (res_mode == 2) {
      sd = dsg[co] * rsqrtf(dsv[co] + BN_EPS);
      dm = dsm[co];
      db = dsb[co];
    }
#pragma unroll
    for (int half = 0; half < 2; half++) {
      int t = t0 + 16 * half + N;
      float val = (half ? acc1[i] : acc0[i]) + bb;
      val = (val - mm) * sc + bt;
      if (res_mode == 1) {
        val += Xres[((size_t)b * cout + co) * TinX + t];
      } else if (res_mode == 2) {
        float r = 0.f;
        const float* Xr = Xres + (size_t)b * cinX * TinX + (size_t)sres * t;
        for (int ci = 0; ci < cinX; ci++) r += dsw[co * cinX + ci] * Xr[(size_t)ci * TinX];
        val += (r - dm) * sd + db;
      }
      out[((size_t)b * cout + co) * Tout + t] = fmaxf(val, 0.f);
    }
  }
}

// ---------------------------------------------------------------------------
// Regressor (1x1 conv 64->4) + clip + band-edge extraction. Reproduces the
// reference's raw reshape (B,4,128) -> (B,128,2,2): flat = 4i+2n+j = c*128+t.
// lo/hi stored as (B, S=128, N=2).
// ---------------------------------------------------------------------------
__global__ void regressor_kernel(const float* __restrict__ y,
                                 const float* __restrict__ wr,
                                 const float* __restrict__ br,
                                 float* __restrict__ lo, float* __restrict__ hi)
{
  int idx = blockIdx.x * blockDim.x + threadIdx.x;
  if (idx >= 2 * 128 * 2) return;
  int n = idx & 1;
  int i = (idx >> 1) & 127;
  int b = idx >> 8;

  float v[2];
#pragma unroll
  for (int j = 0; j < 2; j++) {
    int flat = 4 * i + 2 * n + j;
    int c = flat >> 7;
    int t = flat & 127;
    float acc = br[c];
    const float* yb = y + (size_t)b * 64 * 128 + t;
    for (int ci = 0; ci < 64; ci++) acc += wr[c * 64 + ci] * yb[(size_t)ci * 128];
    acc = acc * (1.f / 6.f) + 0.5f;
    v[j] = fminf(fmaxf(acc, 0.f), 1.f);
  }
  float lov = v[0] * 0.5f;
  float hiv = lov + v[1] * (0.5f - lov);
  lo[((size_t)b * 128 + i) * 2 + n] = lov;
  hi[((size_t)b * 128 + i) * 2 + n] = hiv;
}

// ---------------------------------------------------------------------------
// Final stage: interpolate band edges to T, synthesize the 255-tap band-pass
// filter on the fly, and apply it. Uses
//   hi2*sinc(hi2*m) - lo2*sinc(lo2*m) = (sin(pi*hi2*m) - sin(pi*lo2*m))/(pi*m)
// with both sines advanced by a rotation recurrence (8 FMA/tap, no
// transcendentals in the loop). x windows are staged in LDS.
// ---------------------------------------------------------------------------
__global__ void isolate_kernel(const float* __restrict__ x,
                               const float* __restrict__ lo,
                               const float* __restrict__ hi,
                               float* __restrict__ out, int T, int S)
{
  __shared__ float s_win[NF];
  __shared__ float s_inv[NF];
  __shared__ float s_x[256 + NF - 1];

  int tid = threadIdx.x;
  int t0  = blockIdx.x * 256;
  int b   = blockIdx.y >> 1;
  int n   = blockIdx.y & 1;

  if (tid < NF) {
    float fac = -PI_F + (float)tid * (2.f * PI_F / (float)(NF - 1));
    s_win[tid] = 0.54f + 0.46f * cosf(fac);
    int m = tid - (NF / 2);
    s_inv[tid] = (m == 0) ? 0.f : 1.f / (PI_F * (float)m);
  }
  const float* xb = x + (size_t)b * T;
  for (int j = tid; j < 256 + NF - 1; j += 256) {
    int gi = t0 - (NF / 2) + j;
    s_x[j] = (gi >= 0 && gi < T) ? xb[gi] : 0.f;
  }
  __syncthreads();

  int t = t0 + tid;
  // linear interpolation of band edges (S=128 knots -> T samples)
  float pos = (float)t * ((float)(S - 1) / (float)(T - 1));
  int i0 = (int)floorf(pos);
  if (i0 > S - 1) i0 = S - 1;
  int i1 = i0 + 1; if (i1 > S - 1) i1 = S - 1;
  float fr = pos - (float)i0;
  const float* lob = lo + (size_t)b * S * 2 + n;
  const float* hib = hi + (size_t)b * S * 2 + n;
  float lov = lob[(size_t)i0 * 2] * (1.f - fr) + lob[(size_t)i1 * 2] * fr;
  float hiv = hib[(size_t)i0 * 2] * (1.f - fr) + hib[(size_t)i1 * 2] * fr;
  float lo2 = 2.f * lov, hi2 = 2.f * hiv;

  float thl = PI_F * lo2, thh = PI_F * hi2;
  float sl, cl, sh, ch;
  sincosf(thl * (float)(-(NF / 2)), &sl, &cl);
  sincosf(thh * (float)(-(NF / 2)), &sh, &ch);
  float sdl, cdl, sdh, cdh;
  sincosf(thl, &sdl, &cdl);
  sincosf(thh, &sdh, &cdh);

  float acc = 0.f;
  for (int k = 0; k < NF; k++) {
    float val = (k == NF / 2) ? (hi2 - lo2) : (sh - sl) * s_inv[k];
    acc = fmaf(s_x[tid + k] * s_win[k], val, acc);
    float nsl = fmaf(sl, cdl, cl * sdl);
    cl = fmaf(cl, cdl, -sl * sdl);
    sl = nsl;
    float nsh = fmaf(sh, cdh, ch * sdh);
    ch = fmaf(ch, cdh, -sh * sdh);
    sh = nsh;
  }
  out[((size_t)(b * 2 + n)) * T + t] = acc;
}

// ---------------------------------------------------------------------------
// Host side: unpack the flattened parameter dict (insertion order) and run
// the 10 residual layers (ping-pong in d_ws), regressor, and FIR stage.
// ---------------------------------------------------------------------------
extern "C" void kernel_launch(void* const* d_in, const int* in_sizes, int n_in,
                              void* d_out, int out_size, void* d_ws, size_t ws_size,
                              hipStream_t stream)
{
  (void)in_sizes; (void)n_in; (void)out_size; (void)ws_size;
  static const int CH[11] = {1, 4, 4, 8, 8, 8, 16, 16, 32, 32, 64};
  static const int ST[10] = {3, 2, 2, 1, 3, 2, 2, 1, 2, 2};
  static const int DI[10] = {1, 2, 2, 2, 1, 2, 2, 2, 2, 1};
  const int B = 2, T = 73728;

  const float* x = (const float*)d_in[0];
  int p = 1;
  const float *c1w[10], *c1b[10], *b1g[10], *b1be[10], *b1m[10], *b1v[10];
  const float *c2w[10], *c2b[10], *b2g[10], *b2be[10], *b2m[10], *b2v[10];
  const float *dsw[10], *dsg[10], *dsb[10], *dsm[10], *dsv[10];
  int hasds[10];
  for (int i = 0; i < 10; i++) {
    c1w[i]  = (const float*)d_in[p++]; c1b[i]  = (const float*)d_in[p++];
    b1g[i]  = (const float*)d_in[p++]; b1be[i] = (const float*)d_in[p++];
    b1m[i]  = (const float*)d_in[p++]; b1v[i]  = (const float*)d_in[p++];
    c2w[i]  = (const float*)d_in[p++]; c2b[i]  = (const float*)d_in[p++];
    b2g[i]  = (const float*)d_in[p++]; b2be[i] = (const float*)d_in[p++];
    b2m[i]  = (const float*)d_in[p++]; b2v[i]  = (const float*)d_in[p++];
    hasds[i] = (CH[i] != CH[i + 1]) || (ST[i] > 1);
    if (hasds[i]) {
      dsw[i] = (const float*)d_in[p++];
      dsg[i] = (const float*)d_in[p++]; dsb[i] = (const float*)d_in[p++];
      dsm[i] = (const float*)d_in[p++]; dsv[i] = (const float*)d_in[p++];
    } else {
      dsw[i] = dsg[i] = dsb[i] = dsm[i] = dsv[i] = nullptr;
    }
  }
  const float* rw = (const float*)d_in[p++];
  const float* rb = (const float*)d_in[p++];

  // workspace layout (floats)
  float* ws   = (float*)d_ws;
  float* bufA = ws;                    // up to (2,4,24576)   = 196608
  float* bufB = ws + 196608;           // up to (2,4,12288)   =  98304
  float* bufH = ws + 294912;           // up to (2,4,24576)   = 196608
  float* loB  = ws + 491520;           // (2,128,2) = 512
  float* hiB  = ws + 492032;           // (2,128,2) = 512

  const float* cur = x;
  int Tin = T;
  for (int i = 0; i < 10; i++) {
    int cin = CH[i], cout = CH[i + 1];
    int s = ST[i], d = DI[i], pad = 2 * d;
    int Tout = (Tin - 1) / s + 1;
    float* ob = (i & 1) ? bufB : bufA;

    // --- part 1: h = relu(bn1(conv1(x))) ---
    bool wm1 = ((cout & 15) == 0) && ((Tout & 31) == 0) && ((cin & 3) == 0);
    if (wm1) {
      int tiles = B * (cout / 16) * (Tout / 32);
      conv_layer_wmma<<<tiles, 32, 0, stream>>>(
          cur, cin, Tin, c1w[i], c1b[i], cout, Tout, s, d, pad,
          b1g[i], b1be[i], b1m[i], b1v[i],
          0, nullptr, 0, 0, 0, nullptr, nullptr, nullptr, nullptr, nullptr,
          bufH, B);
    } else {
      int total = B * cout * Tout;
      conv_layer_scalar<<<(total + 255) / 256, 256, 0, stream>>>(
          cur, cin, Tin, c1w[i], c1b[i], cout, Tout, s, d, pad,
          b1g[i], b1be[i], b1m[i], b1v[i],
          0, nullptr, 0, 0, 0, nullptr, nullptr, nullptr, nullptr, nullptr,
          bufH, B);
    }

    // --- part 2: out = relu(bn2(conv2(h)) + res(x)) ---
    int rm = hasds[i] ? 2 : 1;
    bool wm2 = ((cout & 15) == 0) && ((Tout & 31) == 0);
    if (wm2) {
      int tiles = B * (cout / 16) * (Tout / 32);
      conv_layer_wmma<<<tiles, 32, 0, stream>>>(
          bufH, cout, Tout, c2w[i], c2b[i], cout, Tout, 1, d, pad,
          b2g[i], b2be[i], b2m[i], b2v[i],
          rm, cur, cin, Tin, s, dsw[i], dsg[i], dsb[i], dsm[i], dsv[i],
          ob, B);
    } else {
      int total = B * cout * Tout;
      conv_layer_scalar<<<(total + 255) / 256, 256, 0, stream>>>(
          bufH, cout, Tout, c2w[i], c2b[i], cout, Tout, 1, d, pad,
          b2g[i], b2be[i], b2m[i], b2v[i],
          rm, cur, cin, Tin, s, dsw[i], dsg[i], dsb[i], dsm[i], dsv[i],
          ob, B);
    }
    cur = ob;
    Tin = Tout;
  }

  // regressor + band edges (2*128*2 = 512 threads)
  regressor_kernel<<<2, 256, 0, stream>>>(cur, rw, rb, loB, hiB);

  // time-varying FIR: grid (T/256, B*N)
  isolate_kernel<<<dim3(T / 256, 4), 256, 0, stream>>>(
      x, loB, hiB, (float*)d_out, T, 128);
}